// RelNwRegr_36816459661914
// MI455X (gfx1250) — compile-verified
//
#include <hip/hip_runtime.h>

typedef __attribute__((ext_vector_type(2))) float v2f;
typedef __attribute__((ext_vector_type(8))) float v8f;

#define N_Q   8192
#define N_B   8192
#define DIM   32
#define QTILE 16
#define BTILE 16
#define WAVES_PER_WG 8
#define THREADS (WAVES_PER_WG * 32)

__global__ __launch_bounds__(THREADS)
void RelNwRegr_kernel(const float* __restrict__ xb,     // (N_B, 32)
                      const float* __restrict__ yb,     // (N_B,)
                      const float* __restrict__ xq,     // (N_Q, 32)
                      const float* __restrict__ r,      // (N_Q, N_B)
                      const float* __restrict__ sigma_p,
                      const float* __restrict__ rscale_p,
                      float* __restrict__ out)          // (N_Q,)
{
    __shared__ float s_num[QTILE];
    __shared__ float s_den[QTILE];

    const int tid = threadIdx.x;
    if (tid < QTILE) { s_num[tid] = 0.0f; s_den[tid] = 0.0f; }
    __syncthreads();

    const int lane = tid & 31;
    const int wave = tid >> 5;
    const int half = (lane >> 4) & 1;   // 0: lanes 0-15, 1: lanes 16-31
    const int l15  = lane & 15;

    const int q0 = blockIdx.x * QTILE;

    const float inv_sigma = 1.0f / sigma_p[0];
    const float rscale    = rscale_p[0];

    // ---- A fragments: 8 chunks of 16x4 f32 (rows q0..q0+15, K = 0..31) ----
    // ISA layout: lane l15 = row m; half h holds K = 4c+2h, 4c+2h+1.
    v2f a[8];
    {
        const float* abase = xq + (size_t)(q0 + l15) * DIM + 2 * half;
        #pragma unroll
        for (int c = 0; c < 8; ++c) {
            a[c].x = abase[4 * c + 0];
            a[c].y = abase[4 * c + 1];
        }
    }
    // q_sq per row: each lane holds half the row; combine with partner lane (xor 16)
    float qs_part = 0.0f;
    #pragma unroll
    for (int c = 0; c < 8; ++c) qs_part += a[c].x * a[c].x + a[c].y * a[c].y;
    const float q_sq_row = qs_part + __shfl_xor(qs_part, 16, 32);
    // Each lane needs q_sq for the 8 rows it owns in the C/D layout: m = i + 8*half
    float q_sq_i[8];
    #pragma unroll
    for (int i = 0; i < 8; ++i) q_sq_i[i] = __shfl(q_sq_row, i + 8 * half, 32);

    float num[8], den[8];
    #pragma unroll
    for (int i = 0; i < 8; ++i) { num[i] = 0.0f; den[i] = 0.0f; }

    const int btiles = N_B / BTILE;   // 512
    for (int bt = wave; bt < btiles; bt += WAVES_PER_WG) {
        const int b0 = bt * BTILE;

        // ---- B fragments: 8 chunks of 4x16 f32, B[k][n] = xb[b0+n][k] ----
        v2f bm[8];
        const float* bbase = xb + (size_t)(b0 + l15) * DIM + 2 * half;
        #pragma unroll
        for (int c = 0; c < 8; ++c) {
            bm[c].x = bbase[4 * c + 0];
            bm[c].y = bbase[4 * c + 1];
        }
        float bs_part = 0.0f;
        #pragma unroll
        for (int c = 0; c < 8; ++c) bs_part += bm[c].x * bm[c].x + bm[c].y * bm[c].y;
        const float b_sq_n = bs_part + __shfl_xor(bs_part, 16, 32);
        const float y_n = yb[b0 + l15];

        // ---- cross tile: 8 chained f32 WMMAs (K = 32) ----
        v8f acc = {};
        #pragma unroll
        for (int c = 0; c < 8; ++c)
            acc = __builtin_amdgcn_wmma_f32_16x16x4_f32(
                      false, a[c], false, bm[c], (short)0, acc, false, false);

        // ---- stream r in the C/D layout, fused elementwise + accumulate ----
        const float* rbase = r + (size_t)(q0 + 8 * half) * N_B + (b0 + l15);
        // prefetch next tile's r for this wave
        __builtin_prefetch(rbase + (size_t)WAVES_PER_WG * BTILE, 0, 1);

        #pragma unroll
        for (int i = 0; i < 8; ++i) {
            const float rv   = rbase[(size_t)i * N_B];
            float d2 = q_sq_i[i] + b_sq_n - 2.0f * acc[i];
            d2 = fmaxf(d2, 0.0f);
            const float dist = __builtin_sqrtf(d2);
            const float kv   = __expf(fmaf(rscale, rv, -dist * inv_sigma));
            num[i] = fmaf(kv, y_n, num[i]);
            den[i] += kv;
        }
    }

    // ---- reduce over the N (background) dimension: 16 lanes per half ----
    #pragma unroll
    for (int i = 0; i < 8; ++i) {
        #pragma unroll
        for (int m = 1; m < 16; m <<= 1) {
            num[i] += __shfl_xor(num[i], m, 32);
            den[i] += __shfl_xor(den[i], m, 32);
        }
    }
    if (l15 == 0) {
        #pragma unroll
        for (int i = 0; i < 8; ++i) {
            atomicAdd(&s_num[i + 8 * half], num[i]);
            atomicAdd(&s_den[i + 8 * half], den[i]);
        }
    }
    __syncthreads();

    if (tid < QTILE)
        out[q0 + tid] = s_num[tid] / (s_den[tid] + 1e-8f);
}

extern "C" void kernel_launch(void* const* d_in, const int* in_sizes, int n_in,
                              void* d_out, int out_size, void* d_ws, size_t ws_size,
                              hipStream_t stream) {
    const float* xb     = (const float*)d_in[0];  // x_backgnd (8192,32)
    const float* yb     = (const float*)d_in[1];  // y_backgnd (8192,)
    const float* xq     = (const float*)d_in[2];  // x_query   (8192,32)
    const float* r      = (const float*)d_in[3];  // r         (8192,8192)
    const float* sigma  = (const float*)d_in[4];
    const float* rscale = (const float*)d_in[5];
    float* out = (float*)d_out;

    dim3 grid(N_Q / QTILE);   // 512 workgroups, one 16-query tile each
    dim3 block(THREADS);      // 8 waves
    RelNwRegr_kernel<<<grid, block, 0, stream>>>(xb, yb, xq, r, sigma, rscale, out);
}